// Hbv_2_2929167696148
// MI455X (gfx1250) — compile-verified
//
#include <hip/hip_runtime.h>
#include <type_traits>

#define NEARZERO 1e-5f

// x > 0 guaranteed (SM >= NEARZERO, FC >= 50) -> v_log_f32 + v_exp_f32
// (TRANS ops; co-execute with VALU on CDNA5)
__device__ __forceinline__ float fast_pow_pos(float x, float e) {
    return __expf(e * __logf(x));
}

// ---------------------------------------------------------------------------
// HBV step chain shared by both kernels. Mirrors the reference op-for-op.
// ---------------------------------------------------------------------------
struct HbvState { float SNOWPACK, MELTWATER, SM, SUZ, SLZ; };
struct HbvParm  { float BETA, FC, K0, K1, K2, PERCp, UZL, TT, CFMAX,
                        CWH, BETAET, C, invFC, invLPFC, CFRCF; };

__device__ __forceinline__ float hbv_step(HbvState& s, const HbvParm& P,
                                          float pr, float tm, float pe) {
    const float is_rain = (tm >= P.TT) ? 1.0f : 0.0f;
    const float RAIN = pr * is_rain;
    const float SNOW = pr - RAIN;
    s.SNOWPACK += SNOW;
    const float melt = fminf(fmaxf(P.CFMAX * (tm - P.TT), 0.0f), s.SNOWPACK);
    s.MELTWATER += melt;  s.SNOWPACK -= melt;
    const float refreeze = fminf(fmaxf(P.CFRCF * (P.TT - tm), 0.0f), s.MELTWATER);
    s.SNOWPACK += refreeze;  s.MELTWATER -= refreeze;
    const float tosoil = fmaxf(s.MELTWATER - P.CWH * s.SNOWPACK, 0.0f);
    s.MELTWATER -= tosoil;
    const float wet = fminf(fast_pow_pos(s.SM * P.invFC, P.BETA), 1.0f);
    const float recharge = (RAIN + tosoil) * wet;
    s.SM += RAIN + tosoil - recharge;
    const float excess = fmaxf(s.SM - P.FC, 0.0f);
    s.SM -= excess;
    const float evf = fminf(fast_pow_pos(s.SM * P.invLPFC, P.BETAET), 1.0f);
    const float ETact = fminf(s.SM, pe * evf);
    s.SM = fmaxf(s.SM - ETact, NEARZERO);
    const float cap = fminf(s.SLZ, P.C * s.SLZ * (1.0f - fminf(s.SM * P.invFC, 1.0f)));
    s.SM  = fmaxf(s.SM + cap, NEARZERO);
    s.SLZ = fmaxf(s.SLZ - cap, NEARZERO);
    s.SUZ += recharge + excess;
    const float PERC = fminf(s.SUZ, P.PERCp);
    s.SUZ -= PERC;
    const float Q0 = P.K0 * fmaxf(s.SUZ - P.UZL, 0.0f);
    s.SUZ -= Q0;
    const float Q1 = P.K1 * s.SUZ;
    s.SUZ -= Q1;
    s.SLZ = fmaxf(s.SLZ + PERC, 0.0f);
    const float Q2 = P.K2 * s.SLZ;
    s.SLZ -= Q2;
    return Q0 + Q1 + Q2;
}

__device__ __forceinline__ HbvParm hbv_load_params(const float* __restrict__ phy, int g) {
    // Bounds: BETA, FC, K0, K1, K2, LP, PERC, UZL, TT, CFMAX, CFR, CWH, BETAET, C
    const float lo[14] = {1.0f, 50.0f,   0.05f, 0.01f, 0.001f, 0.2f,  0.0f, 0.0f,  -2.5f, 0.5f,  0.0f, 0.0f, 0.3f, 0.0f};
    const float hi[14] = {6.0f, 1000.0f, 0.9f,  0.5f,  0.2f,   1.0f, 10.0f, 100.0f, 2.5f, 10.0f, 0.1f, 0.2f, 5.0f, 1.0f};
    float p[14];
    #pragma unroll
    for (int i = 0; i < 14; ++i)
        p[i] = lo[i] + phy[g * 14 + i] * (hi[i] - lo[i]);
    HbvParm P;
    P.BETA = p[0];  P.FC = p[1];   P.K0 = p[2];    P.K1 = p[3];  P.K2 = p[4];
    P.PERCp = p[6]; P.UZL = p[7];  P.TT = p[8];    P.CFMAX = p[9];
    P.CWH = p[11];  P.BETAET = p[12]; P.C = p[13];
    P.invFC   = 1.0f / p[1];
    P.invLPFC = 1.0f / (p[5] * p[1]);
    P.CFRCF   = p[10] * p[9];
    return P;
}

// ---------------------------------------------------------------------------
// Specialized kernel: T, G compile-time constants. All in-chunk load/store/
// prefetch offsets become instruction immediates; base pointers bumped once
// per chunk.
// ---------------------------------------------------------------------------
template<int T_, int G_>
__global__ __launch_bounds__(256) void hbv_scan_kernel_c(
    const float* __restrict__ precip, const float* __restrict__ temp,
    const float* __restrict__ pet,    const float* __restrict__ phy,
    float* __restrict__ Q)
{
    constexpr int KCH   = 16;           // timesteps per register chunk
    constexpr int NFULL = T_ / KCH;
    constexpr int TAIL  = T_ - NFULL * KCH;

    const int g = blockIdx.x * blockDim.x + threadIdx.x;
    if (g >= G_) return;

    const HbvParm P = hbv_load_params(phy, g);
    HbvState s = {NEARZERO, NEARZERO, NEARZERO, NEARZERO, NEARZERO};

    const float* pp = precip + g;
    const float* tp = temp   + g;
    const float* ep = pet    + g;
    float*       qp = Q      + g;

    // One chunk: 3*N independent NT loads issued as a clause (deep memory
    // pipeline), then N recurrence steps + NT stores. Pointer bump keeps all
    // offsets as compile-time immediates (max (2*KCH-1)*G_*4 = 6.2MB < imm24).
    auto chunk = [&](auto NI) {
        constexpr int N = decltype(NI)::value;
        float pr[N], tm[N], pe[N];
        #pragma unroll
        for (int k = 0; k < N; ++k) {
            pr[k] = __builtin_nontemporal_load(pp + k * G_);
            tm[k] = __builtin_nontemporal_load(tp + k * G_);
            pe[k] = __builtin_nontemporal_load(ep + k * G_);
        }
        #pragma unroll
        for (int k = 0; k < N; ++k)
            __builtin_nontemporal_store(hbv_step(s, P, pr[k], tm[k], pe[k]),
                                        qp + k * G_);
        pp += N * G_;  tp += N * G_;  ep += N * G_;  qp += N * G_;
    };

    for (int c = 0; c < NFULL; ++c) {
        // Prefetch next chunk (16 rows = 3.2MB ahead) into L2:
        // global_prefetch_b8, no counters, stays fully in-bounds.
        if (c + 1 < NFULL) {
            #pragma unroll
            for (int k = 0; k < KCH; ++k) {
                __builtin_prefetch(pp + (KCH + k) * G_, 0, 0);
                __builtin_prefetch(tp + (KCH + k) * G_, 0, 0);
                __builtin_prefetch(ep + (KCH + k) * G_, 0, 0);
            }
        }
        chunk(std::integral_constant<int, KCH>{});
    }
    if constexpr (TAIL > 0)
        chunk(std::integral_constant<int, TAIL>{});
}

// ---------------------------------------------------------------------------
// Generic fallback (runtime T, G) — same algorithm, KCH=8 chunks.
// ---------------------------------------------------------------------------
__global__ __launch_bounds__(256) void hbv_scan_kernel_g(
    const float* __restrict__ precip, const float* __restrict__ temp,
    const float* __restrict__ pet,    const float* __restrict__ phy,
    float* __restrict__ Q, int T, int G)
{
    constexpr int KCH = 8;
    const int g = blockIdx.x * blockDim.x + threadIdx.x;
    if (g >= G) return;

    const HbvParm P = hbv_load_params(phy, g);
    HbvState s = {NEARZERO, NEARZERO, NEARZERO, NEARZERO, NEARZERO};

    const int nfull = T / KCH;
    int t0 = 0;
    for (int c = 0; c < nfull; ++c, t0 += KCH) {
        if (t0 + 2 * KCH <= T) {
            #pragma unroll
            for (int k = 0; k < KCH; ++k) {
                const int off = (t0 + KCH + k) * G + g;
                __builtin_prefetch(precip + off, 0, 0);
                __builtin_prefetch(temp   + off, 0, 0);
                __builtin_prefetch(pet    + off, 0, 0);
            }
        }
        float pr[KCH], tm[KCH], pe[KCH];
        #pragma unroll
        for (int k = 0; k < KCH; ++k) {
            const int idx = (t0 + k) * G + g;
            pr[k] = __builtin_nontemporal_load(precip + idx);
            tm[k] = __builtin_nontemporal_load(temp   + idx);
            pe[k] = __builtin_nontemporal_load(pet    + idx);
        }
        #pragma unroll
        for (int k = 0; k < KCH; ++k)
            __builtin_nontemporal_store(hbv_step(s, P, pr[k], tm[k], pe[k]),
                                        Q + (t0 + k) * G + g);
    }
    for (int t = t0; t < T; ++t) {
        const int idx = t * G + g;
        const float q = hbv_step(s, P,
                                 __builtin_nontemporal_load(precip + idx),
                                 __builtin_nontemporal_load(temp   + idx),
                                 __builtin_nontemporal_load(pet    + idx));
        __builtin_nontemporal_store(q, Q + idx);
    }
}

extern "C" void kernel_launch(void* const* d_in, const int* in_sizes, int n_in,
                              void* d_out, int out_size, void* d_ws, size_t ws_size,
                              hipStream_t stream) {
    const float* precip = (const float*)d_in[0];
    const float* temp   = (const float*)d_in[1];
    const float* pet    = (const float*)d_in[2];
    const float* phy    = (const float*)d_in[3];
    float* Q = (float*)d_out;

    const int G = in_sizes[3] / 14;   // phy_params is [G, 14]
    const int T = in_sizes[0] / G;    // forcings are [T, G]

    const int block = 256;            // 8 wave32 waves per block
    const int grid  = (G + block - 1) / block;

    if (T == 365 && G == 50000) {
        hipLaunchKernelGGL((hbv_scan_kernel_c<365, 50000>), dim3(grid), dim3(block),
                           0, stream, precip, temp, pet, phy, Q);
    } else {
        hipLaunchKernelGGL(hbv_scan_kernel_g, dim3(grid), dim3(block),
                           0, stream, precip, temp, pet, phy, Q, T, G);
    }
}